// GraphProjectUPoint_19799799234728
// MI455X (gfx1250) — compile-verified
//
#include <hip/hip_runtime.h>

#define BATCH 16
#define NPTS  10000
#define CH    256
#define HH    56
#define WW    56

typedef float v4f __attribute__((ext_vector_type(4)));
typedef int   v4i __attribute__((ext_vector_type(4)));

// Pointer types matching the builtin's signature:
//   __builtin_amdgcn_global_load_async_to_lds_b128(v4i __device__*, v4i __shared__*, Ii, Ii)
typedef __attribute__((address_space(1))) v4i* gptr_v4i;
typedef __attribute__((address_space(3))) v4i* lptr_v4i;

// CDNA5 async global->LDS path (ASYNCcnt / s_wait_asynccnt). Guarded so the
// file compiles on any toolchain; fallback is direct coalesced b128 loads.
#if defined(__HIP_DEVICE_COMPILE__) && \
    __has_builtin(__builtin_amdgcn_global_load_async_to_lds_b128) && \
    __has_builtin(__builtin_amdgcn_s_wait_asynccnt)
#define USE_ASYNC 1
#else
#define USE_ASYNC 0
#endif

__device__ __forceinline__ void nt_store4(float* p, v4f v) {
  __builtin_nontemporal_store(v, (v4f*)p);
}

// ---------------------------------------------------------------------------
// Kernel 1: transpose [B,C,H,W] -> [B,H,W,C] so channel gathers are contiguous.
// Tiled 32c x 32w through LDS; NT loads (source read once), regular stores
// (destination stays hot in the 192MB L2 for the gather phase).
// ---------------------------------------------------------------------------
__global__ __launch_bounds__(256) void transpose_kernel(
    const float* __restrict__ in, float* __restrict__ out)
{
  __shared__ float tile[32][33];            // 33 stride: conflict-free (gcd(33,64)=1)
  const int tx = threadIdx.x, ty = threadIdx.y;   // (32, 8)
  const int wt = blockIdx.x * 32;
  const int ct = blockIdx.y * 32;
  const int b  = blockIdx.z / HH;
  const int h  = blockIdx.z % HH;

  #pragma unroll
  for (int i = 0; i < 4; ++i) {
    int c = ct + ty + 8 * i;
    int w = wt + tx;
    if (w < WW)
      tile[ty + 8 * i][tx] =
          __builtin_nontemporal_load(in + (((size_t)b * CH + c) * HH + h) * WW + w);
  }
  __syncthreads();
  #pragma unroll
  for (int i = 0; i < 4; ++i) {
    int w = wt + ty + 8 * i;
    int c = ct + tx;
    if (w < WW)
      out[(((size_t)b * HH + h) * WW + w) * CH + c] = tile[tx][ty + 8 * i];
  }
}

// ---------------------------------------------------------------------------
// Kernel 2: per-point projection -> 4 tap offsets (elements into transposed
// buffer) + 4 bilinear weights. Exactly mirrors the reference formula,
// including the degenerate integer-coordinate case (all weights -> 0).
// ---------------------------------------------------------------------------
__global__ __launch_bounds__(256) void project_kernel(
    const float* __restrict__ verts, const float* __restrict__ proj,
    int4* __restrict__ offs, float4* __restrict__ wts)
{
  const int b = blockIdx.y;
  const int n = blockIdx.x * 256 + threadIdx.x;
  if (n >= NPTS) return;

  const float* M = proj + b * 16;          // uniform per block -> scalar loads
  const float* v = verts + ((size_t)b * NPTS + n) * 3;
  float vx = v[0], vy = v[1], vz = v[2];

  float cx = M[0]  * vx + M[1]  * vy + M[2]  * vz + M[3];
  float cy = M[4]  * vx + M[5]  * vy + M[6]  * vz + M[7];
  float cw = M[12] * vx + M[13] * vy + M[14] * vz + M[15];

  float x = (cx / cw + 1.0f) * 0.5f;
  float y = 1.0f - (cy / cw + 1.0f) * 0.5f;
  float x0 = fminf(fmaxf(x * (float)WW, 0.0f), (float)(WW - 1));
  float y0 = fminf(fmaxf(y * (float)HH, 0.0f), (float)(HH - 1));

  float x1f = floorf(x0), x2f = ceilf(x0);
  float y1f = floorf(y0), y2f = ceilf(y0);
  int x1 = (int)x1f, y1 = (int)y1f;
  int x2 = min((int)x2f, WW - 1);          // clamp keeps address valid; weight is 0 there
  int y2 = min((int)y2f, HH - 1);

  int o11 = ((b * HH + y1) * WW + x1) * CH;
  int o21 = ((b * HH + y1) * WW + x2) * CH;
  int o12 = ((b * HH + y2) * WW + x1) * CH;
  int o22 = ((b * HH + y2) * WW + x2) * CH;

  float dx1 = x0 - x1f, dx2 = x2f - x0;
  float dy1 = y0 - y1f, dy2 = y2f - y0;

  int p = b * NPTS + n;
  offs[p] = make_int4(o11, o21, o12, o22);
  wts[p]  = make_float4(dx2 * dy2, dx1 * dy2, dx2 * dy1, dx1 * dy1);
}

// ---------------------------------------------------------------------------
// Kernel 3: gather + blend. Persistent blocks of (64,4): 4 points per group,
// 64 threads x v4f per point (each wave touches 512B contiguous per tap).
// Async path: double-buffer each group's 4x4KB tap rows into LDS with
// global_load_async_to_lds_b128, overlapping the L2 fetch of group g+stride
// with the blend of group g. NT stores keep the 164MB output out of L2.
// ---------------------------------------------------------------------------
__global__ __launch_bounds__(256) void gather_kernel(
    const float* __restrict__ trans, const int4* __restrict__ offs,
    const float4* __restrict__ wts, float* __restrict__ out, int NG)
{
  const int tx = threadIdx.x;               // 0..63  -> channel chunk
  const int ty = threadIdx.y;               // 0..3   -> point in group
  const int tid = ty * 64 + tx;
  const int stride = gridDim.x;

#if USE_ASYNC
  __shared__ float lbuf[2][4][4][CH];       // [stage][point][tap][channel] = 32 KB
  const int tap = tid >> 6;                 // uniform per wave
  const int chunk = tid & 63;
  int stage = 0;
  const int g0 = blockIdx.x;

  if (g0 < NG) {
    #pragma unroll
    for (int j = 0; j < 4; ++j) {
      int4 o = offs[g0 * 4 + j];
      int base = (tap == 0) ? o.x : (tap == 1) ? o.y : (tap == 2) ? o.z : o.w;
      const float* gp = trans + base + chunk * 4;
      __builtin_amdgcn_global_load_async_to_lds_b128(
          (gptr_v4i)(float*)gp,
          (lptr_v4i)&lbuf[0][j][tap][chunk * 4],
          0, 0);
    }
  }
  for (int g = g0; g < NG; g += stride) {
    int gn = g + stride;
    if (gn < NG) {
      #pragma unroll
      for (int j = 0; j < 4; ++j) {
        int4 o = offs[gn * 4 + j];
        int base = (tap == 0) ? o.x : (tap == 1) ? o.y : (tap == 2) ? o.z : o.w;
        const float* gp = trans + base + chunk * 4;
        __builtin_amdgcn_global_load_async_to_lds_b128(
            (gptr_v4i)(float*)gp,
            (lptr_v4i)&lbuf[stage ^ 1][j][tap][chunk * 4],
            0, 0);
      }
      __builtin_amdgcn_s_wait_asynccnt(4);  // current group's 4 transfers done
    } else {
      __builtin_amdgcn_s_wait_asynccnt(0);
    }
    __syncthreads();                        // all waves' LDS writes visible
    {
      int p = g * 4 + ty;
      float4 w = wts[p];
      int c = tx * 4;
      v4f q11 = *(const v4f*)&lbuf[stage][ty][0][c];
      v4f q21 = *(const v4f*)&lbuf[stage][ty][1][c];
      v4f q12 = *(const v4f*)&lbuf[stage][ty][2][c];
      v4f q22 = *(const v4f*)&lbuf[stage][ty][3][c];
      v4f r = w.x * q11 + w.y * q21 + w.z * q12 + w.w * q22;
      nt_store4(out + (size_t)p * CH + c, r);
    }
    __syncthreads();                        // done reading before stage reuse
    stage ^= 1;
  }
#else
  for (int g = blockIdx.x; g < NG; g += stride) {
    int p = g * 4 + ty;
    int4 o = offs[p];
    float4 w = wts[p];
    int c = tx * 4;
    v4f q11 = *(const v4f*)(trans + o.x + c);
    v4f q21 = *(const v4f*)(trans + o.y + c);
    v4f q12 = *(const v4f*)(trans + o.z + c);
    v4f q22 = *(const v4f*)(trans + o.w + c);
    v4f r = w.x * q11 + w.y * q21 + w.z * q12 + w.w * q22;
    nt_store4(out + (size_t)p * CH + c, r);
  }
#endif
}

// ---------------------------------------------------------------------------
// Fallback if workspace is too small for the transposed layout: fused
// project + gather straight from [B,C,H,W] (uncoalesced but correct).
// ---------------------------------------------------------------------------
__global__ __launch_bounds__(256) void fused_kernel(
    const float* __restrict__ verts, const float* __restrict__ img,
    const float* __restrict__ proj, float* __restrict__ out)
{
  const int p = blockIdx.x * 4 + threadIdx.y;
  const int b = p / NPTS, n = p % NPTS;

  const float* M = proj + b * 16;
  const float* v = verts + ((size_t)b * NPTS + n) * 3;
  float vx = v[0], vy = v[1], vz = v[2];
  float cx = M[0]  * vx + M[1]  * vy + M[2]  * vz + M[3];
  float cy = M[4]  * vx + M[5]  * vy + M[6]  * vz + M[7];
  float cw = M[12] * vx + M[13] * vy + M[14] * vz + M[15];
  float x = (cx / cw + 1.0f) * 0.5f;
  float y = 1.0f - (cy / cw + 1.0f) * 0.5f;
  float x0 = fminf(fmaxf(x * (float)WW, 0.0f), (float)(WW - 1));
  float y0 = fminf(fmaxf(y * (float)HH, 0.0f), (float)(HH - 1));
  float x1f = floorf(x0), x2f = ceilf(x0);
  float y1f = floorf(y0), y2f = ceilf(y0);
  int x1 = (int)x1f, y1 = (int)y1f;
  int x2 = min((int)x2f, WW - 1);
  int y2 = min((int)y2f, HH - 1);
  float w11 = (x2f - x0) * (y2f - y0);
  float w21 = (x0 - x1f) * (y2f - y0);
  float w12 = (x2f - x0) * (y0 - y1f);
  float w22 = (x0 - x1f) * (y0 - y1f);

  const float* base = img + (size_t)b * CH * HH * WW;
  const int c0 = threadIdx.x * 4;
  #pragma unroll
  for (int k = 0; k < 4; ++k) {
    int c = c0 + k;
    const float* pc = base + (size_t)c * HH * WW;
    float q11 = pc[y1 * WW + x1];
    float q21 = pc[y1 * WW + x2];
    float q12 = pc[y2 * WW + x1];
    float q22 = pc[y2 * WW + x2];
    out[(size_t)p * CH + c] = w11 * q11 + w21 * q21 + w12 * q12 + w22 * q22;
  }
}

extern "C" void kernel_launch(void* const* d_in, const int* in_sizes, int n_in,
                              void* d_out, int out_size, void* d_ws, size_t ws_size,
                              hipStream_t stream) {
  (void)in_sizes; (void)n_in; (void)out_size;
  const float* vertices = (const float*)d_in[0];
  const float* img      = (const float*)d_in[1];
  const float* proj     = (const float*)d_in[2];
  float* out = (float*)d_out;

  const size_t transBytes = (size_t)BATCH * HH * WW * CH * sizeof(float);   // 51.4 MB
  const size_t pointBytes = (size_t)BATCH * NPTS * (sizeof(int4) + sizeof(float4));
  const int NG = BATCH * NPTS / 4;   // 40000 groups of 4 points

  if (ws_size >= transBytes + pointBytes) {
    float* trans = (float*)d_ws;
    int4*   offs = (int4*)((char*)d_ws + transBytes);
    float4* wts  = (float4*)((char*)d_ws + transBytes + (size_t)BATCH * NPTS * sizeof(int4));

    transpose_kernel<<<dim3((WW + 31) / 32, CH / 32, BATCH * HH), dim3(32, 8), 0, stream>>>(img, trans);
    project_kernel<<<dim3((NPTS + 255) / 256, BATCH), dim3(256), 0, stream>>>(vertices, proj, offs, wts);

    int blocks = NG < 8192 ? NG : 8192;   // persistent: ~5 groups/block for pipelining
    gather_kernel<<<dim3(blocks), dim3(64, 4), 0, stream>>>(trans, offs, wts, out, NG);
  } else {
    fused_kernel<<<dim3(NG), dim3(64, 4), 0, stream>>>(vertices, img, proj, out);
  }
}